// Dynamic_MultiTeacher4_52699248721948
// MI455X (gfx1250) — compile-verified
//
#include <hip/hip_runtime.h>
#include <stdint.h>
#include <math.h>

// ---------------------------------------------------------------------------
// Dynamic multi-teacher KD loss, MI455X (gfx1250).
// Memory-bound: 160MB fp32, read-once -> ~7us floor @ 23.3 TB/s.
//  * one 256-thread block per row; 5 rows (4 teachers + student) streamed to
//    LDS with CDNA5 async global->LDS b128 loads (ASYNCcnt), non-temporal
//    hint (read-once stream, don't pollute caches).
//  * both reduction passes (top-2/max, exp-sums) run out of LDS.
//  * loss is affine in 1/max_preds, so the global max needs no extra HBM
//    pass: row kernel emits CE_b, D_b, rowmax4_b; finalize kernel reduces.
// ---------------------------------------------------------------------------

#define T_KD   20.0f
#define T_THR  2.0f
#define ALPHA  0.8f
#define B_ROWS 8192
#define C_COLS 1000
#define ROW_PAD 1008      // 1008*4B row pitch in LDS (16B aligned)
#define NT     256        // 8 wave32 per block

__device__ __forceinline__ float wave_max(float v) {
#pragma unroll
    for (int o = 16; o; o >>= 1) v = fmaxf(v, __shfl_xor(v, o, 32));
    return v;
}
__device__ __forceinline__ float wave_sum(float v) {
#pragma unroll
    for (int o = 16; o; o >>= 1) v += __shfl_xor(v, o, 32);
    return v;
}
__device__ __forceinline__ void top2_combine(float& r1, float& r2, float b1, float b2) {
    float n2 = fmaxf(fminf(r1, b1), fmaxf(r2, b2));
    r1 = fmaxf(r1, b1);
    r2 = n2;
}

// simple block reduces for the small finalize kernel
__device__ __forceinline__ float block_sum8(float v, float* red, int tid) {
    v = wave_sum(v);
    if ((tid & 31) == 0) red[tid >> 5] = v;
    __syncthreads();
    float r = 0.f;
#pragma unroll
    for (int i = 0; i < 8; ++i) r += red[i];
    __syncthreads();
    return r;
}
__device__ __forceinline__ float block_max8(float v, float* red, int tid) {
    v = wave_max(v);
    if ((tid & 31) == 0) red[tid >> 5] = v;
    __syncthreads();
    float r = red[0];
#pragma unroll
    for (int i = 1; i < 8; ++i) r = fmaxf(r, red[i]);
    __syncthreads();
    return r;
}

__global__ __launch_bounds__(NT)
void mt4_row_kernel(const float* __restrict__ o1, const float* __restrict__ o2,
                    const float* __restrict__ o3, const float* __restrict__ o4,
                    const float* __restrict__ os, const int*   __restrict__ tgts,
                    float* __restrict__ ce_out, float* __restrict__ d_out_,
                    float* __restrict__ rmax_out) {
    __shared__ __align__(16) float sT[5 * ROW_PAD];   // rows: t0..t3, student
    __shared__ float red[96];                          // packed reduction slab

    const int tid = threadIdx.x;
    const int wid = tid >> 5;
    const int b   = blockIdx.x;
    const size_t rowoff = (size_t)b * C_COLS;

    // ---- async stream 5 rows (4000B each) into LDS: lanes 0..249 copy 16B ---
    const float* srcs[5] = { o1 + rowoff, o2 + rowoff, o3 + rowoff,
                             o4 + rowoff, os + rowoff };
    const uint32_t lds0 = (uint32_t)(uintptr_t)(&sT[0]);  // low 32b of flat = LDS offset
    if (tid < 250) {
#pragma unroll
        for (int r = 0; r < 5; ++r) {
            const float* g = srcs[r] + tid * 4;
            uint32_t l = lds0 + (uint32_t)(r * ROW_PAD * 4 + tid * 16);
            asm volatile("global_load_async_to_lds_b128 %0, %1, off th:TH_LOAD_NT"
                         :: "v"(l), "v"(g) : "memory");
        }
    }
    asm volatile("s_wait_asynccnt 0" ::: "memory");
    __syncthreads();

    // ---- phase A: top-2 per teacher (incl. mimic), student max -------------
    float m1[5], m2[5];
#pragma unroll
    for (int t = 0; t < 5; ++t) { m1[t] = -INFINITY; m2[t] = -INFINITY; }
    float smax = -INFINITY;

    for (int c = tid; c < C_COLS; c += NT) {
        float v0 = sT[0 * ROW_PAD + c], v1 = sT[1 * ROW_PAD + c];
        float v2 = sT[2 * ROW_PAD + c], v3 = sT[3 * ROW_PAD + c];
        float sv = sT[4 * ROW_PAD + c];
        float vm = (v0 + v1 + v2 + v3) * 0.25f;
        float vals[5] = { v0, v1, v2, v3, vm };
#pragma unroll
        for (int t = 0; t < 5; ++t) {
            float v = vals[t];
            if (v > m1[t]) { m2[t] = m1[t]; m1[t] = v; }
            else if (v > m2[t]) { m2[t] = v; }
        }
        smax = fmaxf(smax, sv);
    }

    // wave-level combine, then ONE packed LDS round-trip for all 11 scalars
#pragma unroll
    for (int t = 0; t < 5; ++t) {
#pragma unroll
        for (int o = 16; o; o >>= 1) {
            float b1 = __shfl_xor(m1[t], o, 32), b2 = __shfl_xor(m2[t], o, 32);
            top2_combine(m1[t], m2[t], b1, b2);
        }
    }
    smax = wave_max(smax);
    if ((tid & 31) == 0) {
#pragma unroll
        for (int t = 0; t < 5; ++t) { red[t * 8 + wid] = m1[t]; red[(5 + t) * 8 + wid] = m2[t]; }
        red[80 + wid] = smax;
    }
    __syncthreads();
#pragma unroll
    for (int t = 0; t < 5; ++t) {
        float r1 = red[t * 8], r2 = red[(5 + t) * 8];
#pragma unroll
        for (int i = 1; i < 8; ++i) top2_combine(r1, r2, red[t * 8 + i], red[(5 + t) * 8 + i]);
        m1[t] = r1; m2[t] = r2;
    }
    {
        float r = red[80];
#pragma unroll
        for (int i = 1; i < 8; ++i) r = fmaxf(r, red[80 + i]);
        smax = r;
    }
    __syncthreads();

    // ---- phase B: exp-sums from LDS ----------------------------------------
    const float invT = 1.0f / T_KD;
    float ZT[5] = {0,0,0,0,0}, SS[5] = {0,0,0,0,0};
    float Z1 = 0.f, ZTs = 0.f;

    for (int c = tid; c < C_COLS; c += NT) {
        float v0 = sT[0 * ROW_PAD + c], v1 = sT[1 * ROW_PAD + c];
        float v2 = sT[2 * ROW_PAD + c], v3 = sT[3 * ROW_PAD + c];
        float sv = sT[4 * ROW_PAD + c];
        float vm = (v0 + v1 + v2 + v3) * 0.25f;
        float vals[5] = { v0, v1, v2, v3, vm };
        Z1  += __expf(sv - smax);
        ZTs += __expf((sv - smax) * invT);
#pragma unroll
        for (int t = 0; t < 5; ++t) {
            float e = __expf((vals[t] - m1[t]) * invT);
            ZT[t] += e;
            SS[t] += e * sv;
        }
    }
    // ONE packed LDS round-trip for all 12 sums
    Z1  = wave_sum(Z1);
    ZTs = wave_sum(ZTs);
#pragma unroll
    for (int t = 0; t < 5; ++t) { ZT[t] = wave_sum(ZT[t]); SS[t] = wave_sum(SS[t]); }
    if ((tid & 31) == 0) {
        red[0 * 8 + wid] = Z1;
        red[1 * 8 + wid] = ZTs;
#pragma unroll
        for (int t = 0; t < 5; ++t) {
            red[(2 + t) * 8 + wid] = ZT[t];
            red[(7 + t) * 8 + wid] = SS[t];
        }
    }
    __syncthreads();
    {
        float acc[12];
#pragma unroll
        for (int q = 0; q < 12; ++q) {
            float r = 0.f;
#pragma unroll
            for (int i = 0; i < 8; ++i) r += red[q * 8 + i];
            acc[q] = r;
        }
        Z1 = acc[0]; ZTs = acc[1];
#pragma unroll
        for (int t = 0; t < 5; ++t) { ZT[t] = acc[2 + t]; SS[t] = acc[7 + t]; }
    }
    __syncthreads();

    // ---- per-row scalar math (redundant across threads; lane 0 writes) -----
    const int tg = tgts[b];
    float tv[5];
    tv[0] = sT[0 * ROW_PAD + tg]; tv[1] = sT[1 * ROW_PAD + tg];
    tv[2] = sT[2 * ROW_PAD + tg]; tv[3] = sT[3 * ROW_PAD + tg];
    tv[4] = (tv[0] + tv[1] + tv[2] + tv[3]) * 0.25f;
    const float s_tg = sT[4 * ROW_PAD + tg];

    const float CE     = smax + __logf(Z1) - s_tg;
    const float logZTs = __logf(ZTs);

    float margins[5], kd[5];
#pragma unroll
    for (int t = 0; t < 5; ++t) {
        margins[t] = (tv[t] == m1[t]) ? (m1[t] - m2[t]) : 0.0f;
        kd[t] = T_KD * T_KD * logZTs - T_KD * (SS[t] / ZT[t] - smax);
    }
    // threshold weights: softmax(margins / T_THR)
    float mm = margins[0];
#pragma unroll
    for (int t = 1; t < 5; ++t) mm = fmaxf(mm, margins[t]);
    float em[5], esum = 0.f;
#pragma unroll
    for (int t = 0; t < 5; ++t) { em[t] = __expf((margins[t] - mm) / T_THR); esum += em[t]; }

    float D = 0.f;
#pragma unroll
    for (int t = 0; t < 5; ++t) D += (em[t] / esum) * tv[t] * (kd[t] - CE);

    if (tid == 0) {
        ce_out[b]   = CE;
        d_out_[b]   = D;
        rmax_out[b] = fmaxf(fmaxf(m1[0], m1[1]), fmaxf(m1[2], m1[3]));
    }
}

__global__ __launch_bounds__(NT)
void mt4_finalize_kernel(const float* __restrict__ ce, const float* __restrict__ D,
                         const float* __restrict__ rmax, float* __restrict__ out) {
    __shared__ float red[8];
    const int tid = threadIdx.x;

    float m = -INFINITY;
    for (int i = tid; i < B_ROWS; i += NT) m = fmaxf(m, rmax[i]);
    m = block_max8(m, red, tid);          // = max_preds

    const float coef = ALPHA / m;
    float s = 0.f;
    for (int i = tid; i < B_ROWS; i += NT) s += ce[i] + coef * D[i];
    s = block_sum8(s, red, tid);

    if (tid == 0) out[0] = s * (1.0f / (float)B_ROWS);
}

extern "C" void kernel_launch(void* const* d_in, const int* in_sizes, int n_in,
                              void* d_out, int out_size, void* d_ws, size_t ws_size,
                              hipStream_t stream) {
    (void)in_sizes; (void)n_in; (void)out_size; (void)ws_size;
    const float* o1 = (const float*)d_in[0];
    const float* o2 = (const float*)d_in[1];
    const float* o3 = (const float*)d_in[2];
    const float* o4 = (const float*)d_in[3];
    const float* os = (const float*)d_in[4];
    const int*  tgt = (const int*)  d_in[5];

    float* ws   = (float*)d_ws;           // needs 3*8192*4 = 96KB scratch
    float* ce   = ws;
    float* Dv   = ws + B_ROWS;
    float* rmax = ws + 2 * B_ROWS;

    mt4_row_kernel<<<B_ROWS, NT, 0, stream>>>(o1, o2, o3, o4, os, tgt, ce, Dv, rmax);
    mt4_finalize_kernel<<<1, NT, 0, stream>>>(ce, Dv, rmax, (float*)d_out);
}